// RGGAT_85512798863850
// MI455X (gfx1250) — compile-verified
//
#include <hip/hip_runtime.h>
#include <hip/hip_bf16.h>

// ---------------------------------------------------------------------------
// CDNA5 (gfx1250) fp32 WMMA helpers: V_WMMA_F32_16X16X4_F32
//   A: 16x4 f32  -> v2f per lane  (m = lane&15, k = 2*(lane>>4)+v)
//   B:  4x16 f32 -> v2f per lane  (k = 2*(lane>>4)+v, n = lane&15)
//   C/D: 16x16 f32 -> v8f per lane (row = 8*(lane>>4)+v, col = lane&15)
// ---------------------------------------------------------------------------
typedef float v2f __attribute__((ext_vector_type(2)));
typedef float v8f __attribute__((ext_vector_type(8)));

#define WMMA_F32(a, b, c) \
    __builtin_amdgcn_wmma_f32_16x16x4_f32(false, (a), false, (b), (short)0, (c), false, false)

// EEG region channel groups (compile-time constant tables)
__constant__ int g_chan[32] = {0,2, 1,3, 4,6,8, 5,7,9, 10,11,12,13,
                               14,16,18, 15,17,19, 20,22,24, 21,23,25,
                               26,27,28,29,30,31};
__constant__ int g_roff[11] = {0,2,4,7,10,14,17,20,23,26,32};

#define NB    2048   // batch
#define NF    1024   // initial feature dim
#define ND    64     // embedding dim
#define NNODE 10     // region nodes
#define HC1   1024   // gat1 heads*channels
#define NC2   256    // gat2 channels

struct EncArgs {
    const float* x;
    const float* w1[10];
    const float* b1[10];
    const float* w2[10];
    const float* b2[10];
    float*       nodes;   // [B, 10, 64]
};

// ---------------------------------------------------------------------------
// Kernel 1: fused region encoder.
//   block = (m-tile of 128 samples, region r)
//   GEMM1: rf[128, Kr] @ w1[Kr,128]  -> relu(+b1) tile in LDS
//   GEMM2: tile[128,128] @ w2[128,64] (+b2) -> nodes
// ---------------------------------------------------------------------------
__global__ __launch_bounds__(256) void enc_kernel(EncArgs args) {
    const int r    = blockIdx.y;
    const int m0   = blockIdx.x * 128;
    const int tid  = threadIdx.x;
    const int lane = tid & 31;
    const int wave = tid >> 5;
    const int l15  = lane & 15;
    const int hi2  = (lane >> 4) * 2;
    const int Kr   = (g_roff[r + 1] - g_roff[r]) * NF;
    const int cb   = g_roff[r];

    __shared__ float As[128 * 33];    // A tile, padded stride
    __shared__ float Bs[32 * 128];    // B tile
    __shared__ float Hs[128 * 132];   // relu(GEMM1) tile, padded stride
    __shared__ float Ws2[128 * 68];   // w2, padded stride

    const float* w1 = args.w1[r];

    v8f zero = {};
    v8f acc[8];
#pragma unroll
    for (int i = 0; i < 8; ++i) acc[i] = zero;

    for (int kt = 0; kt < Kr; kt += 32) {
        // 32-col tile always lies inside one channel's 1024-float span
        const int   ch  = g_chan[cb + (kt >> 10)];
        const int   f0  = kt & 1023;
        const float* xs = args.x + (size_t)ch * NF + f0;
#pragma unroll
        for (int p = 0; p < 4; ++p) {           // 128x32 A tile: 4 float4 / thread
            int idx = tid + p * 256;            // 0..1023
            int row = idx >> 3;
            int c4  = (idx & 7) * 4;
            const float4 v = *(const float4*)(xs + (size_t)(m0 + row) * (32 * NF) + c4);
            float* d = &As[row * 33 + c4];
            d[0] = v.x; d[1] = v.y; d[2] = v.z; d[3] = v.w;
        }
#pragma unroll
        for (int p = 0; p < 4; ++p) {           // 32x128 B tile
            int idx = tid + p * 256;
            int row = idx >> 5;
            int c4  = (idx & 31) * 4;
            *(float4*)&Bs[row * 128 + c4] = *(const float4*)(w1 + (size_t)(kt + row) * 128 + c4);
        }
        __syncthreads();
#pragma unroll
        for (int ks = 0; ks < 32; ks += 4) {
            v2f a;
            a.x = As[(wave * 16 + l15) * 33 + ks + hi2 + 0];
            a.y = As[(wave * 16 + l15) * 33 + ks + hi2 + 1];
#pragma unroll
            for (int nt = 0; nt < 8; ++nt) {
                v2f b;
                b.x = Bs[(ks + hi2 + 0) * 128 + nt * 16 + l15];
                b.y = Bs[(ks + hi2 + 1) * 128 + nt * 16 + l15];
                acc[nt] = WMMA_F32(a, b, acc[nt]);
            }
        }
        __syncthreads();
    }

    // relu(+b1) -> Hs
    {
        const float* b1 = args.b1[r];
#pragma unroll
        for (int nt = 0; nt < 8; ++nt) {
            int   col  = nt * 16 + l15;
            float bias = b1[col];
#pragma unroll
            for (int v = 0; v < 8; ++v) {
                int   row = wave * 16 + (lane >> 4) * 8 + v;
                float t   = acc[nt][v] + bias;
                Hs[row * 132 + col] = t > 0.f ? t : 0.f;
            }
        }
    }
    // stage w2 [128,64] into LDS (padded stride 68)
    {
        const float* w2 = args.w2[r];
#pragma unroll
        for (int p = 0; p < 8; ++p) {
            int idx = tid + p * 256;            // 0..2047 float4s
            int row = idx >> 4;
            int c4  = (idx & 15) * 4;
            const float4 v = *(const float4*)(w2 + row * 64 + c4);
            float* d = &Ws2[row * 68 + c4];
            d[0] = v.x; d[1] = v.y; d[2] = v.z; d[3] = v.w;
        }
    }
    __syncthreads();

    v8f acc2[4];
#pragma unroll
    for (int i = 0; i < 4; ++i) acc2[i] = zero;
#pragma unroll 4
    for (int k = 0; k < 128; k += 4) {
        v2f a;
        a.x = Hs[(wave * 16 + l15) * 132 + k + hi2 + 0];
        a.y = Hs[(wave * 16 + l15) * 132 + k + hi2 + 1];
#pragma unroll
        for (int nt = 0; nt < 4; ++nt) {
            v2f b;
            b.x = Ws2[(k + hi2 + 0) * 68 + nt * 16 + l15];
            b.y = Ws2[(k + hi2 + 1) * 68 + nt * 16 + l15];
            acc2[nt] = WMMA_F32(a, b, acc2[nt]);
        }
    }
    {
        const float* b2 = args.b2[r];
#pragma unroll
        for (int nt = 0; nt < 4; ++nt) {
            int   d    = nt * 16 + l15;
            float bias = b2[d];
#pragma unroll
            for (int v = 0; v < 8; ++v) {
                int row = wave * 16 + (lane >> 4) * 8 + v;
                args.nodes[((size_t)(m0 + row) * NNODE + r) * ND + d] = acc2[nt][v] + bias;
            }
        }
    }
}

// ---------------------------------------------------------------------------
// Kernel 2/4: generic fp32 WMMA GEMM  C[M,N] = A[M,K] @ B[K,N]
//   requires M%128==0, K%32==0, N%128==0
// ---------------------------------------------------------------------------
__global__ __launch_bounds__(256) void gemm_wmma(const float* __restrict__ A,
                                                 const float* __restrict__ Bm,
                                                 float* __restrict__ C,
                                                 int M, int K, int N) {
    const int m0   = blockIdx.x * 128;
    const int n0   = blockIdx.y * 128;
    const int tid  = threadIdx.x;
    const int lane = tid & 31;
    const int wave = tid >> 5;
    const int l15  = lane & 15;
    const int hi2  = (lane >> 4) * 2;

    __shared__ float As[128 * 33];
    __shared__ float Bs[32 * 128];

    v8f zero = {};
    v8f acc[8];
#pragma unroll
    for (int i = 0; i < 8; ++i) acc[i] = zero;

    for (int kt = 0; kt < K; kt += 32) {
#pragma unroll
        for (int p = 0; p < 4; ++p) {
            int idx = tid + p * 256;
            int row = idx >> 3;
            int c4  = (idx & 7) * 4;
            const float4 v = *(const float4*)(A + (size_t)(m0 + row) * K + kt + c4);
            float* d = &As[row * 33 + c4];
            d[0] = v.x; d[1] = v.y; d[2] = v.z; d[3] = v.w;
        }
#pragma unroll
        for (int p = 0; p < 4; ++p) {
            int idx = tid + p * 256;
            int row = idx >> 5;
            int c4  = (idx & 31) * 4;
            *(float4*)&Bs[row * 128 + c4] =
                *(const float4*)(Bm + (size_t)(kt + row) * N + n0 + c4);
            if (kt + 32 < K)   // speculative prefetch of next B panel -> global_prefetch
                __builtin_prefetch(Bm + (size_t)(kt + 32 + row) * N + n0 + c4, 0, 1);
        }
        __syncthreads();
#pragma unroll
        for (int ks = 0; ks < 32; ks += 4) {
            v2f a;
            a.x = As[(wave * 16 + l15) * 33 + ks + hi2 + 0];
            a.y = As[(wave * 16 + l15) * 33 + ks + hi2 + 1];
#pragma unroll
            for (int nt = 0; nt < 8; ++nt) {
                v2f b;
                b.x = Bs[(ks + hi2 + 0) * 128 + nt * 16 + l15];
                b.y = Bs[(ks + hi2 + 1) * 128 + nt * 16 + l15];
                acc[nt] = WMMA_F32(a, b, acc[nt]);
            }
        }
        __syncthreads();
    }
#pragma unroll
    for (int nt = 0; nt < 8; ++nt) {
        int col = n0 + nt * 16 + l15;
#pragma unroll
        for (int v = 0; v < 8; ++v) {
            int row = m0 + wave * 16 + (lane >> 4) * 8 + v;
            C[(size_t)row * N + col] = acc[nt][v];
        }
    }
}

// ---------------------------------------------------------------------------
// Kernel 3: GAT1 attention (8 heads, 10 nodes, fully connected), elu(+bias)
//   one block per sample
// ---------------------------------------------------------------------------
__global__ __launch_bounds__(256) void gat1_attn(const float* __restrict__ hp1,
                                                 const float* __restrict__ att_src,
                                                 const float* __restrict__ att_dst,
                                                 const float* __restrict__ bias,
                                                 float* __restrict__ h1) {
    const int b   = blockIdx.x;
    const int tid = threadIdx.x;
    __shared__ float hp[NNODE * HC1];          // 40 KB
    __shared__ float s_src[NNODE][8];
    __shared__ float s_dst[NNODE][8];
    __shared__ float alpha[NNODE][NNODE][8];   // [dst][src][head]

    const float* src = hp1 + (size_t)b * (NNODE * HC1);
#pragma unroll
    for (int p = 0; p < 10; ++p) {             // 2560 float4s / 256 threads
        int idx = tid + p * 256;
        *(float4*)&hp[idx * 4] = *(const float4*)(src + idx * 4);
    }
    __syncthreads();

    if (tid < 80) {                            // (node, head) score dots
        int   n = tid / 8, h = tid % 8;
        float s = 0.f, d = 0.f;
        for (int c = 0; c < 128; ++c) {
            float v = hp[n * HC1 + h * 128 + c];
            s += v * att_src[h * 128 + c];
            d += v * att_dst[h * 128 + c];
        }
        s_src[n][h] = s;
        s_dst[n][h] = d;
    }
    __syncthreads();

    if (tid < 80) {                            // softmax over sources per (dst, head)
        int   i = tid / 8, h = tid % 8;
        float e[NNODE], mx = -1e30f;
        for (int j = 0; j < NNODE; ++j) {
            float t = s_dst[i][h] + s_src[j][h];
            t = t > 0.f ? t : 0.2f * t;        // leaky_relu(0.2)
            e[j] = t;
            mx = fmaxf(mx, t);
        }
        float sum = 0.f;
        for (int j = 0; j < NNODE; ++j) { e[j] = __expf(e[j] - mx); sum += e[j]; }
        float inv = 1.f / sum;
        for (int j = 0; j < NNODE; ++j) alpha[i][j][h] = e[j] * inv;
    }
    __syncthreads();

    float* dst = h1 + (size_t)b * (NNODE * HC1);
    for (int i = 0; i < NNODE; ++i) {
        for (int c = tid; c < HC1; c += 256) {
            int   h = c >> 7;
            float s = 0.f;
#pragma unroll
            for (int j = 0; j < NNODE; ++j) s += alpha[i][j][h] * hp[j * HC1 + c];
            s += bias[c];
            dst[i * HC1 + c] = s > 0.f ? s : (__expf(s) - 1.f);   // elu
        }
    }
}

// ---------------------------------------------------------------------------
// Kernel 5: GAT2 attention (1 head) + bias + mean pool over 10 nodes
//   pooled[c] = sum_j (sum_i alpha[i][j] / 10) * hp2[j,c] + bias[c]
// ---------------------------------------------------------------------------
__global__ __launch_bounds__(256) void gat2_attn_pool(const float* __restrict__ hp2,
                                                      const float* __restrict__ att_src,
                                                      const float* __restrict__ att_dst,
                                                      const float* __restrict__ bias,
                                                      float* __restrict__ pooled) {
    const int b   = blockIdx.x;
    const int tid = threadIdx.x;
    __shared__ float hp[NNODE * NC2];          // 10 KB
    __shared__ float s_src[NNODE], s_dst[NNODE];
    __shared__ float alpha[NNODE][NNODE];
    __shared__ float wcol[NNODE];

    const float* src = hp2 + (size_t)b * (NNODE * NC2);
    for (int idx = tid; idx < NNODE * NC2; idx += 256) hp[idx] = src[idx];
    __syncthreads();

    if (tid < NNODE) {
        float s = 0.f, d = 0.f;
        for (int c = 0; c < NC2; ++c) {
            float v = hp[tid * NC2 + c];
            s += v * att_src[c];
            d += v * att_dst[c];
        }
        s_src[tid] = s;
        s_dst[tid] = d;
    }
    __syncthreads();
    if (tid < NNODE) {
        int   i = tid;
        float e[NNODE], mx = -1e30f;
        for (int j = 0; j < NNODE; ++j) {
            float t = s_dst[i] + s_src[j];
            t = t > 0.f ? t : 0.2f * t;
            e[j] = t;
            mx = fmaxf(mx, t);
        }
        float sum = 0.f;
        for (int j = 0; j < NNODE; ++j) { e[j] = __expf(e[j] - mx); sum += e[j]; }
        float inv = 1.f / sum;
        for (int j = 0; j < NNODE; ++j) alpha[i][j] = e[j] * inv;
    }
    __syncthreads();
    if (tid < NNODE) {
        float s = 0.f;
        for (int i = 0; i < NNODE; ++i) s += alpha[i][tid];
        wcol[tid] = s * 0.1f;                  // fold mean-pool into source weights
    }
    __syncthreads();
    {
        float s = bias[tid];                   // mean of broadcast bias == bias
#pragma unroll
        for (int j = 0; j < NNODE; ++j) s += wcol[j] * hp[j * NC2 + tid];
        pooled[(size_t)b * NC2 + tid] = s;
    }
}

// ---------------------------------------------------------------------------
// Kernel 6: classifier  relu(pooled@w1+b1)@w2+b2 -> logits [B,2]
// ---------------------------------------------------------------------------
__global__ __launch_bounds__(256) void cls_kernel(const float* __restrict__ pooled,
                                                  const float* __restrict__ w1,
                                                  const float* __restrict__ b1,
                                                  const float* __restrict__ w2,
                                                  const float* __restrict__ b2,
                                                  float* __restrict__ out) {
    const int m0   = blockIdx.x * 128;
    const int tid  = threadIdx.x;
    const int lane = tid & 31;
    const int wave = tid >> 5;
    const int l15  = lane & 15;
    const int hi2  = (lane >> 4) * 2;

    __shared__ float As[128 * 33];
    __shared__ float Bs[32 * 128];
    __shared__ float Hs[128 * 132];
    __shared__ float W2s[128 * 2];

    W2s[tid] = w2[tid];                        // 256 elements

    v8f zero = {};
    v8f acc[8];
#pragma unroll
    for (int i = 0; i < 8; ++i) acc[i] = zero;

    for (int kt = 0; kt < 256; kt += 32) {
#pragma unroll
        for (int p = 0; p < 4; ++p) {
            int idx = tid + p * 256;
            int row = idx >> 3;
            int c4  = (idx & 7) * 4;
            const float4 v = *(const float4*)(pooled + (size_t)(m0 + row) * 256 + kt + c4);
            float* d = &As[row * 33 + c4];
            d[0] = v.x; d[1] = v.y; d[2] = v.z; d[3] = v.w;
        }
#pragma unroll
        for (int p = 0; p < 4; ++p) {
            int idx = tid + p * 256;
            int row = idx >> 5;
            int c4  = (idx & 31) * 4;
            *(float4*)&Bs[row * 128 + c4] = *(const float4*)(w1 + (size_t)(kt + row) * 128 + c4);
        }
        __syncthreads();
#pragma unroll
        for (int ks = 0; ks < 32; ks += 4) {
            v2f a;
            a.x = As[(wave * 16 + l15) * 33 + ks + hi2 + 0];
            a.y = As[(wave * 16 + l15) * 33 + ks + hi2 + 1];
#pragma unroll
            for (int nt = 0; nt < 8; ++nt) {
                v2f b;
                b.x = Bs[(ks + hi2 + 0) * 128 + nt * 16 + l15];
                b.y = Bs[(ks + hi2 + 1) * 128 + nt * 16 + l15];
                acc[nt] = WMMA_F32(a, b, acc[nt]);
            }
        }
        __syncthreads();
    }
#pragma unroll
    for (int nt = 0; nt < 8; ++nt) {
        int   col  = nt * 16 + l15;
        float bias = b1[col];
#pragma unroll
        for (int v = 0; v < 8; ++v) {
            int   row = wave * 16 + (lane >> 4) * 8 + v;
            float t   = acc[nt][v] + bias;
            Hs[row * 132 + col] = t > 0.f ? t : 0.f;
        }
    }
    __syncthreads();
    {
        int   row = tid >> 1;                  // 128 rows x 2 logits
        int   col = tid & 1;
        float s   = b2[col];
        for (int k = 0; k < 128; ++k) s += Hs[row * 132 + k] * W2s[k * 2 + col];
        out[(size_t)(m0 + row) * 2 + col] = s;
    }
}

// ---------------------------------------------------------------------------
// launcher
// ---------------------------------------------------------------------------
extern "C" void kernel_launch(void* const* d_in, const int* in_sizes, int n_in,
                              void* d_out, int out_size, void* d_ws, size_t ws_size,
                              hipStream_t stream) {
    (void)in_sizes; (void)n_in; (void)out_size; (void)ws_size;

    EncArgs ea;
    ea.x = (const float*)d_in[0];
    for (int r = 0; r < 10; ++r) {
        ea.w1[r] = (const float*)d_in[1 + 4 * r + 0];
        ea.b1[r] = (const float*)d_in[1 + 4 * r + 1];
        ea.w2[r] = (const float*)d_in[1 + 4 * r + 2];
        ea.b2[r] = (const float*)d_in[1 + 4 * r + 3];
    }
    // d_in[41]=Wq, d_in[42]=Wk: adjacency is computed-then-deleted in the
    // reference (GATConv ignores edge weights) -> no output effect, skipped.
    const float* gat1_w    = (const float*)d_in[43];
    const float* gat1_asrc = (const float*)d_in[44];
    const float* gat1_adst = (const float*)d_in[45];
    const float* gat1_b    = (const float*)d_in[46];
    const float* gat2_w    = (const float*)d_in[47];
    const float* gat2_asrc = (const float*)d_in[48];
    const float* gat2_adst = (const float*)d_in[49];
    const float* gat2_b    = (const float*)d_in[50];
    const float* cls_w1    = (const float*)d_in[51];
    const float* cls_b1    = (const float*)d_in[52];
    const float* cls_w2    = (const float*)d_in[53];
    const float* cls_b2    = (const float*)d_in[54];

    float* ws     = (float*)d_ws;
    float* nodes  = ws;                                   // [2048,10,64]
    float* hp1    = nodes + (size_t)NB * NNODE * ND;      // [20480,1024]
    float* h1     = hp1 + (size_t)NB * NNODE * HC1;       // [20480,1024]
    float* hp2    = h1 + (size_t)NB * NNODE * HC1;        // [20480,256]
    float* pooled = hp2 + (size_t)NB * NNODE * NC2;       // [2048,256]
    ea.nodes = nodes;

    enc_kernel<<<dim3(NB / 128, 10), 256, 0, stream>>>(ea);
    gemm_wmma<<<dim3(NB * NNODE / 128, HC1 / 128), 256, 0, stream>>>(
        nodes, gat1_w, hp1, NB * NNODE, ND, HC1);
    gat1_attn<<<NB, 256, 0, stream>>>(hp1, gat1_asrc, gat1_adst, gat1_b, h1);
    gemm_wmma<<<dim3(NB * NNODE / 128, NC2 / 128), 256, 0, stream>>>(
        h1, gat2_w, hp2, NB * NNODE, HC1, NC2);
    gat2_attn_pool<<<NB, 256, 0, stream>>>(hp2, gat2_asrc, gat2_adst, gat2_b, pooled);
    cls_kernel<<<NB / 128, 256, 0, stream>>>(pooled, cls_w1, cls_b1, cls_w2, cls_b2,
                                             (float*)d_out);
}